// BiSNN_7593502179411
// MI455X (gfx1250) — compile-verified
//
#include <hip/hip_runtime.h>

#define BB   64
#define CCH  64
#define HH   112
#define WW   112
#define HID  256
#define TT   4
#define PLANE  (HH*WW)      // 12544
#define PLANE4 (PLANE/4)    // 3136
#define ROW4   (WW/4)       // 28
#define BN_EPS 1e-5f

typedef float v8f __attribute__((ext_vector_type(8)));
typedef float v2f __attribute__((ext_vector_type(2)));

// ---------------- Kernel 1: global average pool over each (b,c) plane ----------------
__global__ void gap_kernel(const float* __restrict__ x, float* __restrict__ xfeat) {
    const int bc = blockIdx.x;
    const float4* p = (const float4*)(x + (size_t)bc * PLANE);
    float s = 0.f;
    for (int i = threadIdx.x; i < PLANE4; i += 256) {
        float4 v = p[i];
        s += v.x + v.y + v.z + v.w;
    }
    for (int off = 16; off > 0; off >>= 1) s += __shfl_xor(s, off, 32);
    __shared__ float red[8];
    const int lane = threadIdx.x & 31, wv = threadIdx.x >> 5;
    if (lane == 0) red[wv] = s;
    __syncthreads();
    if (threadIdx.x == 0) {
        float t = 0.f;
        for (int i = 0; i < 8; ++i) t += red[i];
        xfeat[bc] = t * (1.0f / (float)PLANE);
    }
}

// ---------------- Kernel 2: GEMMs (WMMA f32) + BN1/ReLU + PLIF + table build ----------------
// one block, 256 threads = 8 waves (wave32)
__global__ void core_kernel(const float* __restrict__ xfeat,
                            const float* __restrict__ w_in,
                            const float* __restrict__ g1, const float* __restrict__ b1,
                            const float* __restrict__ m1, const float* __restrict__ var1,
                            const float* __restrict__ decay_p, const float* __restrict__ vth_p,
                            const float* __restrict__ w_out, const float* __restrict__ conv_w,
                            const float* __restrict__ g2, const float* __restrict__ b2,
                            const float* __restrict__ m2, const float* __restrict__ var2,
                            const float* __restrict__ scale,
                            float* __restrict__ table) {
    __shared__ float hls[BB * HID];   // 64 KB: h_vec, then reused for binary spikes

    const int tid  = threadIdx.x;
    const int lane = tid & 31;
    const int wv   = tid >> 5;
    const int hf   = lane >> 4;   // 0 or 1 (upper half of wave)
    const int l15  = lane & 15;

    // ---- GEMM1: h(64x256) = xfeat(64x64) @ w_in^T(64x256) ; B[k][n] = w_in[n*64+k] ----
    for (int t = wv * 8; t < wv * 8 + 8; ++t) {       // 64 tiles of 16x16, 8 per wave
        const int tm = t >> 4, tn = t & 15;
        v8f acc = {0.f,0.f,0.f,0.f,0.f,0.f,0.f,0.f};
        for (int k0 = 0; k0 < CCH; k0 += 4) {
            v2f a, b;
            const int ar = (tm * 16 + l15) * CCH + k0 + 2 * hf;  // A: row-major [M][K]
            a[0] = xfeat[ar];  a[1] = xfeat[ar + 1];
            const int br = (tn * 16 + l15) * CCH + k0 + 2 * hf;  // B^T = w_in: [N][K]
            b[0] = w_in[br];   b[1] = w_in[br + 1];
            acc = __builtin_amdgcn_wmma_f32_16x16x4_f32(false, a, false, b,
                                                        (short)0, acc, false, false);
        }
        for (int v = 0; v < 8; ++v) {                 // D layout: M = v + 8*hf, N = l15
            const int m = tm * 16 + v + 8 * hf;
            const int n = tn * 16 + l15;
            const float rs = rsqrtf(var1[n] + BN_EPS) * g1[n];
            const float val = (acc[v] - m1[n]) * rs + b1[n];
            hls[m * HID + n] = fmaxf(val, 0.f);       // BN1 + ReLU
        }
    }
    __syncthreads();

    // ---- PLIF recurrence over T steps; each wave owns 8 rows, each lane 8 hidden cols ----
    const float d   = 1.0f / (1.0f + __expf(-decay_p[0]));
    const float vth = vth_p[0];
    for (int b = wv * 8; b < wv * 8 + 8; ++b) {
        float hv[8], ms[8], ss[8];
        for (int u = 0; u < 8; ++u) {
            hv[u] = hls[b * HID + lane * 8 + u];
            ms[u] = 0.f; ss[u] = 0.f;
        }
        for (int step = 0; step < TT; ++step) {
            float mem[8]; float pa = 0.f;
            for (int u = 0; u < 8; ++u) {
                mem[u] = d * (ms[u] - vth * ss[u]) + hv[u];
                pa += fabsf(mem[u]);
            }
            for (int off = 16; off > 0; off >>= 1) pa += __shfl_xor(pa, off, 32);
            const float inv = 1.0f / (pa * (1.0f / (float)HID) + 1e-6f);
            for (int u = 0; u < 8; ++u) {
                mem[u] *= inv;
                ms[u] = mem[u];
                ss[u] = (mem[u] >= vth) ? 1.f : 0.f;
            }
        }
        for (int u = 0; u < 8; ++u)                   // binary = 2*spike - 1 (reuse hls)
            hls[b * HID + lane * 8 + u] = 2.f * ss[u] - 1.f;
    }
    __syncthreads();

    // ---- GEMM2: o(64x64) = binary(64x256) @ w_out^T(256x64); B[k][n] = w_out[n*256+k] ----
    for (int t = wv * 2; t < wv * 2 + 2; ++t) {       // 16 tiles, 2 per wave
        const int tm = t >> 2, tn = t & 3;
        v8f acc = {0.f,0.f,0.f,0.f,0.f,0.f,0.f,0.f};
        for (int k0 = 0; k0 < HID; k0 += 4) {
            v2f a, b;
            const int ar = (tm * 16 + l15) * HID + k0 + 2 * hf;
            a[0] = hls[ar];    a[1] = hls[ar + 1];
            const int br = (tn * 16 + l15) * HID + k0 + 2 * hf;
            b[0] = w_out[br];  b[1] = w_out[br + 1];
            acc = __builtin_amdgcn_wmma_f32_16x16x4_f32(false, a, false, b,
                                                        (short)0, acc, false, false);
        }
        // mod_map -> 9-case conv/BN2/ReLU/tanh table per (b, c)
        for (int v = 0; v < 8; ++v) {
            const int bi = tm * 16 + v + 8 * hf;
            const int c  = tn * 16 + l15;
            const float mod = 1.0f + 0.5f * tanhf(scale[c] * acc[v]);
            const float* w = conv_w + c * 9;
            float cs[3][3];                           // per-ky column sums (L, mid, R)
            for (int ky = 0; ky < 3; ++ky) {
                const float w0 = w[ky*3+0], w1 = w[ky*3+1], w2 = w[ky*3+2];
                cs[ky][0] = w1 + w2;                  // left edge (kx=0 clipped)
                cs[ky][1] = w0 + w1 + w2;             // interior
                cs[ky][2] = w0 + w1;                  // right edge (kx=2 clipped)
            }
            const float rs = rsqrtf(var2[c] + BN_EPS) * g2[c];
            const float sh = b2[c] - m2[c] * rs;
            float* tp = table + ((size_t)bi * CCH + c) * 9;
            for (int rc = 0; rc < 3; ++rc) {
                for (int cc = 0; cc < 3; ++cc) {
                    float S;
                    if (rc == 0)      S = cs[1][cc] + cs[2][cc];             // top row
                    else if (rc == 1) S = cs[0][cc] + cs[1][cc] + cs[2][cc]; // interior
                    else              S = cs[0][cc] + cs[1][cc];             // bottom row
                    const float y = mod * S;
                    const float z = fmaxf(y * rs + sh, 0.f);
                    tp[rc * 3 + cc] = 1.0f + 0.25f * tanhf(z);
                }
            }
        }
    }
}

// ---------------- Kernel 3: stream the 9-case values out as float4 stores ----------------
__global__ void broadcast_kernel(const float* __restrict__ table, float* __restrict__ out) {
    const int bc = blockIdx.x;
    __shared__ float t9[9];
    if (threadIdx.x < 9) t9[threadIdx.x] = table[(size_t)bc * 9 + threadIdx.x];
    __syncthreads();
    float4* po = (float4*)(out + (size_t)bc * PLANE);
    for (int i = threadIdx.x; i < PLANE4; i += 256) {
        const int row = i / ROW4;
        const int q   = i - row * ROW4;
        const int rc  = (row == 0) ? 0 : ((row == HH - 1) ? 2 : 1);
        const float v0 = t9[rc * 3 + 0], v1 = t9[rc * 3 + 1], v2 = t9[rc * 3 + 2];
        float4 o;
        o.x = (q == 0)        ? v0 : v1;
        o.y = v1;
        o.z = v1;
        o.w = (q == ROW4 - 1) ? v2 : v1;
        po[i] = o;
    }
}

extern "C" void kernel_launch(void* const* d_in, const int* in_sizes, int n_in,
                              void* d_out, int out_size, void* d_ws, size_t ws_size,
                              hipStream_t stream) {
    const float* x       = (const float*)d_in[0];
    const float* w_in    = (const float*)d_in[1];
    const float* g1      = (const float*)d_in[2];
    const float* b1      = (const float*)d_in[3];
    const float* m1      = (const float*)d_in[4];
    const float* var1    = (const float*)d_in[5];
    const float* decay_p = (const float*)d_in[6];
    const float* vth_p   = (const float*)d_in[7];
    const float* w_out   = (const float*)d_in[8];
    const float* conv_w  = (const float*)d_in[9];
    const float* g2      = (const float*)d_in[10];
    const float* b2      = (const float*)d_in[11];
    const float* m2      = (const float*)d_in[12];
    const float* var2    = (const float*)d_in[13];
    const float* scale   = (const float*)d_in[14];

    float* ws    = (float*)d_ws;
    float* xfeat = ws;            // 4096 floats
    float* table = ws + 4096;     // 64*64*9 = 36864 floats
    float* out   = (float*)d_out;

    gap_kernel<<<BB * CCH, 256, 0, stream>>>(x, xfeat);
    core_kernel<<<1, 256, 0, stream>>>(xfeat, w_in, g1, b1, m1, var1,
                                       decay_p, vth_p, w_out, conv_w,
                                       g2, b2, m2, var2, scale, table);
    broadcast_kernel<<<BB * CCH, 256, 0, stream>>>(table, out);
}